// localWeight_25752623907304
// MI455X (gfx1250) — compile-verified
//
#include <hip/hip_runtime.h>
#include <hip/hip_bf16.h>

#define NN 50000
#define EE 800000

typedef __attribute__((ext_vector_type(2))) float v2f;
typedef __attribute__((ext_vector_type(8))) float v8f;

// ---------------------------------------------------------------- degree
__global__ __launch_bounds__(256) void k_deg(const int* __restrict__ dst,
                                             float* __restrict__ deg, int ne) {
  int e = blockIdx.x * 256 + threadIdx.x;
  if (e < ne) atomicAdd(&deg[dst[e]], 1.0f);
}

__global__ __launch_bounds__(256) void k_dinv(float* __restrict__ deg, int n) {
  int i = blockIdx.x * 256 + threadIdx.x;
  if (i < n) deg[i] = rsqrtf(deg[i] + 1.0f);
}

// ---------------------------------------------------------------- GEMM
// Out[N,64] = H[N,64] @ W[64,64] using V_WMMA_F32_16X16X4_F32.
// One wave computes a 16-row x 64-col strip (4 accum tiles), K=64 in 16 steps.
// A frag: lane<16 -> (M=lane, K=k0..k0+1), lane>=16 -> (M=lane-16, K=k0+2..3).
// B frag mirrors A with N in place of M. C/D: VGPR r, lane<16 -> (M=r,N=lane),
// lane>=16 -> (M=r+8, N=lane-16).
__global__ __launch_bounds__(256) void k_gemm64(const float* __restrict__ H,
                                                const float* __restrict__ W,
                                                float* __restrict__ Out,
                                                int nrows) {
  __shared__ float Wl[64 * 64];
  for (int i = threadIdx.x; i < 64 * 64; i += 256) Wl[i] = W[i];
  __syncthreads();

  const int lane = threadIdx.x & 31;
  const int wave = threadIdx.x >> 5;
  const long row0 = ((long)blockIdx.x * 8 + wave) * 16;
  if (row0 >= nrows) return;              // whole-wave uniform: EXEC stays all-1s

  const int l15 = lane & 15;
  const int kh  = (lane >> 4) * 2;        // 0 or 2 (K sub-pair per lane half)
  const float* __restrict__ arow = H + (row0 + l15) * 64;

  v8f acc0 = {}, acc1 = {}, acc2 = {}, acc3 = {};
#pragma unroll
  for (int k0 = 0; k0 < 64; k0 += 4) {
    v2f a = *(const v2f*)(arow + k0 + kh);            // global_load_b64
    const float* wr0 = &Wl[(k0 + kh) * 64 + l15];     // LDS, conflict-free
    const float* wr1 = wr0 + 64;
    v2f b0; b0.x = wr0[0];  b0.y = wr1[0];
    v2f b1; b1.x = wr0[16]; b1.y = wr1[16];
    v2f b2; b2.x = wr0[32]; b2.y = wr1[32];
    v2f b3; b3.x = wr0[48]; b3.y = wr1[48];
    acc0 = __builtin_amdgcn_wmma_f32_16x16x4_f32(false, a, false, b0, (short)0, acc0, false, false);
    acc1 = __builtin_amdgcn_wmma_f32_16x16x4_f32(false, a, false, b1, (short)0, acc1, false, false);
    acc2 = __builtin_amdgcn_wmma_f32_16x16x4_f32(false, a, false, b2, (short)0, acc2, false, false);
    acc3 = __builtin_amdgcn_wmma_f32_16x16x4_f32(false, a, false, b3, (short)0, acc3, false, false);
  }

  const int mo = (lane >> 4) * 8;
  float* outp = Out + (row0 + mo) * 64 + l15;
#pragma unroll
  for (int r = 0; r < 8; ++r) {
    outp[r * 64 + 0]  = acc0[r];
    outp[r * 64 + 16] = acc1[r];
    outp[r * 64 + 32] = acc2[r];
    outp[r * 64 + 48] = acc3[r];
  }
}

// ---------------------------------------------------------------- edge scatter
// One wave per edge: 32 lanes x 2 floats = 64 features. Gathered reads hit L2
// (node table = 12.8MB << 192MB L2); scatter via fp32 L2 atomics.
__global__ __launch_bounds__(256) void k_edge64(const float* __restrict__ Hlin,
                                                const int* __restrict__ src,
                                                const int* __restrict__ dst,
                                                const float* __restrict__ dinv,
                                                float* __restrict__ Agg, int ne) {
  const int lane = threadIdx.x & 31;
  const int e = blockIdx.x * 8 + (threadIdx.x >> 5);
  if (e >= ne) return;
  const int s = src[e], d = dst[e];
  const float nrm = dinv[s] * dinv[d];
  v2f v = *(const v2f*)(Hlin + (long)s * 64 + lane * 2);
  float* p = Agg + (long)d * 64 + lane * 2;
  atomicAdd(p,     v.x * nrm);
  atomicAdd(p + 1, v.y * nrm);
}

// ---------------------------------------------------------------- self + bias + sigmoid
__global__ __launch_bounds__(256) void k_post(const float* __restrict__ Agg,
                                              const float* __restrict__ Hlin,
                                              const float* __restrict__ dinv,
                                              const float* __restrict__ bias,
                                              float* __restrict__ Hout, int n) {
  long i = (long)blockIdx.x * 256 + threadIdx.x;
  if (i >= (long)n * 64) return;
  int node = (int)(i >> 6);
  int f = (int)(i & 63);
  float dn = dinv[node];
  float x = Agg[i] + dn * dn * Hlin[i] + bias[f];
  Hout[i] = 1.0f / (1.0f + __expf(-x));
}

// ---------------------------------------------------------------- final layer (64 -> 1)
__global__ __launch_bounds__(256) void k_gemv(const float* __restrict__ H,
                                              const float* __restrict__ W2,
                                              float* __restrict__ out, int n) {
  int i = blockIdx.x * 256 + threadIdx.x;
  if (i >= n) return;
  const float* h = H + (long)i * 64;
  float s = 0.0f;
#pragma unroll
  for (int k = 0; k < 64; ++k) s += h[k] * W2[k];
  out[i] = s;
}

__global__ __launch_bounds__(256) void k_edge1(const float* __restrict__ hlin2,
                                               const int* __restrict__ src,
                                               const int* __restrict__ dst,
                                               const float* __restrict__ dinv,
                                               float* __restrict__ agg2, int ne) {
  int e = blockIdx.x * 256 + threadIdx.x;
  if (e >= ne) return;
  int s = src[e], d = dst[e];
  atomicAdd(&agg2[d], hlin2[s] * dinv[s] * dinv[d]);
}

__global__ __launch_bounds__(256) void k_final(const float* __restrict__ agg2,
                                               const float* __restrict__ hlin2,
                                               const float* __restrict__ dinv,
                                               const float* __restrict__ b2,
                                               float* __restrict__ out, int n) {
  int i = blockIdx.x * 256 + threadIdx.x;
  if (i >= n) return;
  float dn = dinv[i];
  float x = agg2[i] + dn * dn * hlin2[i] + b2[0];   // TEMP == 1.0
  out[i] = 1.0f / (1.0f + __expf(-x)) + 1e-6f;
}

// ----------------------------------------------------------------
extern "C" void kernel_launch(void* const* d_in, const int* in_sizes, int n_in,
                              void* d_out, int out_size, void* d_ws, size_t ws_size,
                              hipStream_t stream) {
  const float* x  = (const float*)d_in[0];
  const int*   ei = (const int*)d_in[1];
  const float* W0 = (const float*)d_in[3];
  const float* b0 = (const float*)d_in[4];
  const float* W1 = (const float*)d_in[5];
  const float* b1 = (const float*)d_in[6];
  const float* W2 = (const float*)d_in[7];
  const float* b2 = (const float*)d_in[8];
  const int* src = ei;        // edge_index[0]
  const int* dst = ei + EE;   // edge_index[1]

  // workspace layout (floats): dinv | hlin | agg | hcur | hlin2 | agg2  (~39 MB)
  float* ws    = (float*)d_ws;
  float* dinv  = ws;
  float* hlin  = dinv + NN;
  float* agg   = hlin + (size_t)NN * 64;
  float* hcur  = agg  + (size_t)NN * 64;
  float* hlin2 = hcur + (size_t)NN * 64;
  float* agg2  = hlin2 + NN;

  dim3 blk(256);
  const int gE    = (EE + 255) / 256;
  const int gN    = (NN + 255) / 256;
  const int gNF   = (NN * 64 + 255) / 256;
  const int gGemm = ((NN / 16) + 7) / 8;   // 3125 row-tiles, 8 waves/block
  const int gEdgW = (EE + 7) / 8;          // one wave per edge

  // normalization
  hipMemsetAsync(dinv, 0, NN * sizeof(float), stream);
  k_deg<<<gE, blk, 0, stream>>>(dst, dinv, EE);
  k_dinv<<<gN, blk, 0, stream>>>(dinv, NN);

  // layer 0
  k_gemm64<<<gGemm, blk, 0, stream>>>(x, W0, hlin, NN);
  hipMemsetAsync(agg, 0, (size_t)NN * 64 * sizeof(float), stream);
  k_edge64<<<gEdgW, blk, 0, stream>>>(hlin, src, dst, dinv, agg, EE);
  k_post<<<gNF, blk, 0, stream>>>(agg, hlin, dinv, b0, hcur, NN);

  // layer 1
  k_gemm64<<<gGemm, blk, 0, stream>>>(hcur, W1, hlin, NN);
  hipMemsetAsync(agg, 0, (size_t)NN * 64 * sizeof(float), stream);
  k_edge64<<<gEdgW, blk, 0, stream>>>(hlin, src, dst, dinv, agg, EE);
  k_post<<<gNF, blk, 0, stream>>>(agg, hlin, dinv, b1, hcur, NN);

  // layer 2 (64 -> 1)
  k_gemv<<<gN, blk, 0, stream>>>(hcur, W2, hlin2, NN);
  hipMemsetAsync(agg2, 0, NN * sizeof(float), stream);
  k_edge1<<<gE, blk, 0, stream>>>(hlin2, src, dst, dinv, agg2, EE);
  k_final<<<gN, blk, 0, stream>>>(agg2, hlin2, dinv, b2, (float*)d_out, NN);
}